// RRSVM_Module_15238543966803
// MI455X (gfx1250) — compile-verified
//
#include <hip/hip_runtime.h>
#include <stdint.h>

#define B_    16
#define C_    128
#define HO    56
#define WO    56
#define NPIX  (HO * WO)       // 3136 = 98 waves * 32
#define TROWS 58              // 56 data rows + top/bottom zero pad rows
#define NT    256             // 8 waves
#define KK    9
#define NVEC4 (NPIX / 4)      // 784 float4 per plane

// Order-preserving float -> u32 bijection (ascending), then pack rank so that
// u64 "max-first" sorting == stable descending argsort (value desc, index asc).
__device__ __forceinline__ unsigned long long mkkey(float f, int k) {
    int b = __float_as_int(f);
    unsigned u = (unsigned)(b ^ ((b >> 31) | 0x80000000));
    return ((unsigned long long)u << 32) | (unsigned)(15 - k);
}

__global__ __launch_bounds__(NT) void rrsvm_rankpool_kernel(
    const float* __restrict__ x,     // [B, C, 56, 56]
    const float* __restrict__ s,     // [C, 9]
    float* __restrict__ out,         // [B, C, 56, 56]
    float* __restrict__ idxf)        // [B, C, 56, 56, 9] (indices as float)
{
    __shared__ __align__(16) float tile[TROWS * WO];        // rows 0 & 57 = pad zeros
    __shared__ __align__(16) float stage[2][NT / 32][32 * KK]; // double-buffered drain
    __shared__ float wts[KK];

    const int plane = blockIdx.x;            // b*C + c
    const int c     = plane % C_;
    const int tid   = threadIdx.x;
    const int lane  = tid & 31;
    const int wave  = tid >> 5;

    if (tid < KK) wts[tid] = s[c * KK + tid];
    if (tid < WO) {                           // zero only the two pad rows
        tile[tid] = 0.0f;
        tile[57 * WO + tid] = 0.0f;
    }

    // Async b128 DMA: plane is one contiguous 12544B block -> 16B-aligned
    // interior of the tile starting at row 1 (float offset 56, byte 224).
    {
        const unsigned long long gbase =
            (unsigned long long)(uintptr_t)(x + (size_t)plane * NPIX);
        const unsigned lbase = (unsigned)(uintptr_t)&tile[WO];
        for (int f = tid; f < NVEC4; f += NT) {
            unsigned la = lbase + (unsigned)(f * 16);
            unsigned long long ga = gbase + (unsigned long long)(f * 16);
            asm volatile("global_load_async_to_lds_b128 %0, %1, off"
                         :: "v"(la), "v"(ga) : "memory");
        }
    }
    asm volatile("s_wait_asynccnt 0" ::: "memory");
    __syncthreads();

    const size_t obase = (size_t)plane * NPIX;

    // Max-first compare-exchange on packed keys: v_cmp_lt_u64 + cndmasks.
#define CE(i, j)                                                      \
    {                                                                 \
        const bool sw_ = key[i] < key[j];                             \
        const unsigned long long ta_ = sw_ ? key[j] : key[i];         \
        const unsigned long long tb_ = sw_ ? key[i] : key[j];         \
        key[i] = ta_; key[j] = tb_;                                   \
    }

    int it = 0;
    for (int pw = wave * 32; pw < NPIX; pw += NT, ++it) {
        const int p = pw + lane;
        const int h = p / WO, w = p - h * WO;

        // Window gather; tile row = input row + 1, columns clamped + zeroed.
        const bool lv = (w > 0), rv = (w < WO - 1);
        const int cl = lv ? w - 1 : 0;
        const int cr = rv ? w + 1 : WO - 1;
        unsigned long long key[KK];
#pragma unroll
        for (int kh = 0; kh < 3; ++kh) {
            const float* row = &tile[(h + kh) * WO];
            float fl = row[cl], fm = row[w], fr = row[cr];
            fl = lv ? fl : 0.0f;
            fr = rv ? fr : 0.0f;
            key[kh * 3 + 0] = mkkey(fl, kh * 3 + 0);
            key[kh * 3 + 1] = mkkey(fm, kh * 3 + 1);
            key[kh * 3 + 2] = mkkey(fr, kh * 3 + 2);
        }

        // Optimal 9-input sorting network: 25 CE, depth 7 (Dobbelaere).
        CE(0, 3) CE(1, 7) CE(2, 5) CE(4, 8)
        CE(0, 7) CE(2, 4) CE(3, 8) CE(5, 6)
        CE(0, 2) CE(1, 3) CE(4, 5) CE(7, 8)
        CE(1, 4) CE(3, 6) CE(5, 7)
        CE(0, 1) CE(2, 4) CE(3, 5) CE(6, 8)
        CE(2, 3) CE(4, 5) CE(6, 7)
        CE(1, 2) CE(3, 4) CE(5, 6)

        // Unpack: inverse bijection for values, low nibble for ranks.
        float acc = 0.0f;
        float idxv[KK];
#pragma unroll
        for (int k = 0; k < KK; ++k) {
            unsigned u = (unsigned)(key[k] >> 32);
            int m = (int)(~u) >> 31;                  // 0 if MSB(u)==1, else -1
            float v = __int_as_float((int)(u ^ (unsigned)(m | 0x80000000)));
            acc = fmaf(v, wts[k], acc);
            idxv[k] = (float)(15 - (int)((unsigned)key[k] & 15u));
        }
        out[obase + p] = acc;                         // coalesced b32 store

        // Double-buffered index drain: wait only for the buffer reused now
        // (stores complete in order; <=3 outstanding == last iteration's).
        asm volatile("s_wait_asynccnt 3" ::: "memory");
        float* sp = &stage[it & 1][wave][0];
#pragma unroll
        for (int k = 0; k < KK; ++k)                  // lane*9+k: 9 coprime 64
            sp[lane * KK + k] = idxv[k];
        asm volatile("s_wait_dscnt 0" ::: "memory");  // visible to async engine

        // 1152B wave region, 16B-aligned both sides -> 3x async b128 stores.
        const unsigned sla = (unsigned)(uintptr_t)sp + (unsigned)(lane * 16);
        const unsigned long long sga =
            (unsigned long long)(uintptr_t)(idxf + (obase + (size_t)pw) * KK) +
            (unsigned long long)(lane * 16);
        asm volatile("global_store_async_from_lds_b128 %0, %1, off"
                     :: "v"(sga), "v"(sla) : "memory");
        asm volatile("global_store_async_from_lds_b128 %0, %1, off offset:512"
                     :: "v"(sga), "v"(sla) : "memory");
        if (lane < 8)
            asm volatile("global_store_async_from_lds_b128 %0, %1, off offset:1024"
                         :: "v"(sga), "v"(sla) : "memory");
    }
#undef CE

    asm volatile("s_wait_asynccnt 0" ::: "memory");   // drain before LDS dealloc
}

extern "C" void kernel_launch(void* const* d_in, const int* in_sizes, int n_in,
                              void* d_out, int out_size, void* d_ws, size_t ws_size,
                              hipStream_t stream) {
    (void)in_sizes; (void)n_in; (void)out_size; (void)d_ws; (void)ws_size;
    const float* x = (const float*)d_in[0];   // [16,128,56,56] f32
    const float* s = (const float*)d_in[1];   // [128,3,3] f32
    float* out  = (float*)d_out;                              // output 0: pooled
    float* idxf = out + (size_t)B_ * C_ * HO * WO;            // output 1: indices
    rrsvm_rankpool_kernel<<<B_ * C_, NT, 0, stream>>>(x, s, out, idxf);
}